// SwitchLinear_16183436771716
// MI455X (gfx1250) — compile-verified
//
#include <hip/hip_runtime.h>
#include <math.h>

// ---------------------------------------------------------------------------
// Switch-MoE (top-1) for MI455X / gfx1250, fp32 end-to-end.
//   K1 router   : logits + softmax + argmax  (bandwidth bound, VALU)
//   K2 scan     : sequential-order position scan, builds gather lists,
//                 handles dropped-token passthrough (single WG, ballot scan)
//   K3 GEMM     : grouped GEMM via V_WMMA_F32_16X16X4_F32 with LDS
//                 double-buffering driven by GLOBAL_LOAD_ASYNC_TO_LDS_B128
//                 (ASYNCcnt-paced), fused bias + route-prob scale + scatter.
// Shapes compile-time specialized for D=512, E=16 (the reference config).
// ---------------------------------------------------------------------------

typedef __attribute__((ext_vector_type(2))) float v2f;
typedef __attribute__((ext_vector_type(8))) float v8f;

// ---------------------------------------------------------------------------
// Kernel 1: router. 16 tokens / block, thread (t) = (token t>>4, expert t&15).
// ---------------------------------------------------------------------------
__global__ __launch_bounds__(256) void router_kernel(
    const float* __restrict__ x, const float* __restrict__ sw,
    const float* __restrict__ sb, int* __restrict__ routes,
    float* __restrict__ probs, int N) {
  constexpr int D = 512, E = 16, TOK = 16, LDX = D + 8;  // stride 520
  __shared__ float xs[TOK][LDX];
  const int tid = threadIdx.x;
  const long long tok0 = (long long)blockIdx.x * TOK;

#pragma unroll
  for (int i = 0; i < 8; ++i) {
    int flat = (i * 256 + tid) * 4;  // 0..8191
    int r = flat >> 9;
    int c = flat & 511;
    float4 v = *(const float4*)(x + (tok0 + r) * D + c);
    *(float4*)&xs[r][c] = v;
  }
  __syncthreads();

  const int tl = tid >> 4;   // token within block
  const int e = tid & 15;    // expert
  const float* __restrict__ wrow = sw + e * D;
  float acc = 0.f;
#pragma unroll 8
  for (int d = 0; d < D; d += 4) {
    float4 a = *(const float4*)&xs[tl][d];
    float4 w = *(const float4*)(wrow + d);
    acc += a.x * w.x + a.y * w.y + a.z * w.z + a.w * w.w;
  }
  float logit = acc + sb[e];

  float m = logit;
#pragma unroll
  for (int off = 8; off >= 1; off >>= 1) m = fmaxf(m, __shfl_xor(m, off, 32));
  float p = expf(logit - m);
  float s = p;
#pragma unroll
  for (int off = 8; off >= 1; off >>= 1) s += __shfl_xor(s, off, 32);
  int idx = (logit == m) ? e : E;  // first-index tie-break like argmax
#pragma unroll
  for (int off = 8; off >= 1; off >>= 1) idx = min(idx, __shfl_xor(idx, off, 32));

  if (e == 0) {
    long long n = tok0 + tl;
    routes[n] = idx;
    probs[n] = 1.0f / s;  // max softmax prob = exp(0)/sum
  }
}

// ---------------------------------------------------------------------------
// Kernel 2: order-preserving per-expert position scan (single WG, 32 waves).
// ---------------------------------------------------------------------------
__global__ __launch_bounds__(1024) void scan_dispatch_kernel(
    const int* __restrict__ routes, const float* __restrict__ probs,
    const float* __restrict__ x, int* __restrict__ tok_idx,
    int* __restrict__ cnt, float* __restrict__ out, int N, int capacity) {
  constexpr int E = 16, D = 512;
  __shared__ int waveCnt[32][E];
  __shared__ int wavePref[32][E];
  __shared__ int base[E];
  const int tid = threadIdx.x;
  const int lane = tid & 31;
  const int wid = tid >> 5;
  if (tid < E) base[tid] = 0;
  __syncthreads();

  for (int chunk = 0; chunk < N; chunk += 1024) {
    const int i = chunk + tid;
    const int r = (i < N) ? routes[i] : -1;
    int rank = 0;
#pragma unroll
    for (int e = 0; e < E; ++e) {
      unsigned mask = (unsigned)__ballot(r == e);  // wave32: low 32 bits
      if (lane == 0) waveCnt[wid][e] = __popc(mask);
      if (r == e) rank = __popc(mask & ((1u << lane) - 1u));
    }
    __syncthreads();
    if (tid < E) {
      int run = base[tid];
#pragma unroll 8
      for (int w = 0; w < 32; ++w) {
        int c = waveCnt[w][tid];
        wavePref[w][tid] = run;
        run += c;
      }
      base[tid] = run;
    }
    __syncthreads();
    if (i < N) {
      const int pos = wavePref[wid][r] + rank;
      if (pos < capacity) {
        tok_idx[r * capacity + pos] = i;
      } else {  // dropped token: out = x * route_prob
        const float p = probs[i];
        const float4* __restrict__ src = (const float4*)(x + (long long)i * D);
        float4* __restrict__ dst = (float4*)(out + (long long)i * D);
        for (int d = 0; d < D / 4; ++d) {
          float4 v = src[d];
          v.x *= p; v.y *= p; v.z *= p; v.w *= p;
          dst[d] = v;
        }
      }
    }
    __syncthreads();
  }
  if (tid < E) cnt[tid] = min(base[tid], capacity);
}

// ---------------------------------------------------------------------------
// Kernel 3: grouped expert GEMM, fp32 WMMA, async double-buffered LDS.
// Block tile 128(M) x 64(N), K slabs of 32. 8 waves in 4x2 grid, each wave a
// 32x32 register tile (4 x v8f). LDS stride = 36 floats: 36L mod 64 spans 16
// distinct multiples of 4; lane-halves (+2) fill odd pairs -> all 64 banks,
// conflict-free b64 fragment reads. Global->LDS staging uses
// GLOBAL_LOAD_ASYNC_TO_LDS_B128 (6 per wave per slab): issue slab j+1, then
// s_wait_asynccnt 6 retires exactly slab j (async loads complete in order),
// barrier makes it block-visible. OOB gather rows clamp to token 0 (their C
// rows are never stored), keeping EXEC full for unconditional async issue.
// ---------------------------------------------------------------------------
__global__ __launch_bounds__(256) void expert_gemm_kernel(
    const float* __restrict__ x, const float* __restrict__ ew,
    const float* __restrict__ eb, const float* __restrict__ probs,
    const int* __restrict__ tok_idx, const int* __restrict__ cnt,
    float* __restrict__ out, int capacity) {
  constexpr int D = 512;
  constexpr int TM = 128, TN = 64, TK = 32;
  constexpr int LDT = TK + 4;  // 36 floats (144 B rows, 16B aligned)
  __shared__ float As[2][TM][LDT];
  __shared__ float Bs[2][TN][LDT];
  __shared__ int toks[TM];

  const int e = blockIdx.z;
  const int count = cnt[e];
  const int m0 = blockIdx.y * TM;
  if (m0 >= count) return;  // empty tile for this expert
  const int n0 = blockIdx.x * TN;

  const int tid = threadIdx.x;
  if (tid < TM) {
    int gm = m0 + tid;
    toks[tid] = (gm < count) ? tok_idx[e * capacity + gm] : -1;
  }
  __syncthreads();

  const int lane = tid & 31;
  const int wid = tid >> 5;
  const int wm = wid & 3;            // 0..3: 32-row sub-tile
  const int wn = wid >> 2;           // 0..1: 32-col sub-tile
  const int hk = (lane >> 4) << 1;   // K half: 0 or 2
  const int l16 = lane & 15;

  const float* __restrict__ W = ew + (long long)e * D * D;

  // Async stage one K-slab (A: 4 x b128/thread, B: 2 x b128/thread).
  auto stage_async = [&](int buf, int kb) {
#pragma unroll
    for (int i = 0; i < 4; ++i) {
      int flat = (i * 256 + tid) * 4;  // 0..4095
      int r = flat >> 5;               // row 0..127
      int c = flat & 31;
      int t = toks[r];
      int tc = t < 0 ? 0 : t;          // clamp: garbage rows never stored
      unsigned lds = (unsigned)(size_t)&As[buf][r][c];
      unsigned off = (unsigned)(((tc << 9) + kb + c) * 4);
      asm volatile("global_load_async_to_lds_b128 %0, %1, %2"
                   :: "v"(lds), "v"(off), "s"(x) : "memory");
    }
#pragma unroll
    for (int i = 0; i < 2; ++i) {
      int flat = (i * 256 + tid) * 4;  // 0..2047
      int r = flat >> 5;               // row 0..63
      int c = flat & 31;
      unsigned lds = (unsigned)(size_t)&Bs[buf][r][c];
      unsigned off = (unsigned)(((unsigned)(n0 + r) * D + kb + c) * 4);
      asm volatile("global_load_async_to_lds_b128 %0, %1, %2"
                   :: "v"(lds), "v"(off), "s"(W) : "memory");
    }
  };

  v8f acc00 = {0, 0, 0, 0, 0, 0, 0, 0};
  v8f acc01 = {0, 0, 0, 0, 0, 0, 0, 0};
  v8f acc10 = {0, 0, 0, 0, 0, 0, 0, 0};
  v8f acc11 = {0, 0, 0, 0, 0, 0, 0, 0};

  stage_async(0, 0);  // prologue: fill buffer 0

  constexpr int NSLAB = D / TK;  // 16
  for (int j = 0; j < NSLAB; ++j) {
    const int buf = j & 1;
    if (j + 1 < NSLAB) {
      stage_async(buf ^ 1, (j + 1) * TK);            // overlap next slab
      asm volatile("s_wait_asynccnt 0x6" ::: "memory");  // cur slab resident
    } else {
      asm volatile("s_wait_asynccnt 0x0" ::: "memory");
    }
    __syncthreads();  // all waves' async writes for `buf` visible

#pragma unroll
    for (int ks = 0; ks < TK; ks += 4) {
      // A frag: lane m = l16, VGPR pair = K{hk, hk+1}   (16x4 layout)
      v2f a0 = *(const v2f*)&As[buf][wm * 32 + l16][ks + hk];
      v2f a1 = *(const v2f*)&As[buf][wm * 32 + l16 + 16][ks + hk];
      // B frag: lane n = l16, VGPR pair = K{hk, hk+1}   (4x16 layout, W^T)
      v2f b0 = *(const v2f*)&Bs[buf][wn * 32 + l16][ks + hk];
      v2f b1 = *(const v2f*)&Bs[buf][wn * 32 + l16 + 16][ks + hk];
      acc00 = __builtin_amdgcn_wmma_f32_16x16x4_f32(false, a0, false, b0,
                                                    (short)0, acc00, false, false);
      acc01 = __builtin_amdgcn_wmma_f32_16x16x4_f32(false, a0, false, b1,
                                                    (short)0, acc01, false, false);
      acc10 = __builtin_amdgcn_wmma_f32_16x16x4_f32(false, a1, false, b0,
                                                    (short)0, acc10, false, false);
      acc11 = __builtin_amdgcn_wmma_f32_16x16x4_f32(false, a1, false, b1,
                                                    (short)0, acc11, false, false);
    }
    __syncthreads();  // reads of `buf` done before it becomes the next target
  }

  // Epilogue: C layout = 8 VGPRs, lanes 0-15 -> M=vr, lanes 16-31 -> M=vr+8,
  // N = lane&15. Apply bias + route-prob scale, scatter by token id.
  const int col = l16;
  const int mh = (lane >> 4) << 3;  // 0 or 8
  auto epi = [&](const v8f& a, int i, int j) {
    const int nG = n0 + wn * 32 + j * 16 + col;
    const float bias = eb[e * D + nG];
#pragma unroll
    for (int vr = 0; vr < 8; ++vr) {
      int rl = wm * 32 + i * 16 + mh + vr;
      int t = toks[rl];
      if (t >= 0) {
        out[(long long)t * D + nG] = (a[vr] + bias) * probs[t];
      }
    }
  };
  epi(acc00, 0, 0);
  epi(acc01, 0, 1);
  epi(acc10, 1, 0);
  epi(acc11, 1, 1);
}

// ---------------------------------------------------------------------------
// Host launcher.
// ---------------------------------------------------------------------------
extern "C" void kernel_launch(void* const* d_in, const int* in_sizes, int n_in,
                              void* d_out, int out_size, void* d_ws,
                              size_t ws_size, hipStream_t stream) {
  const float* x = (const float*)d_in[0];        // [B,S,D]
  const float* sw = (const float*)d_in[1];       // [E,D]
  const float* sb = (const float*)d_in[2];       // [E]
  const float* ew = (const float*)d_in[3];       // [E,D,D]
  const float* eb = (const float*)d_in[4];       // [E,D]
  float* out = (float*)d_out;

  const int E = in_sizes[2];            // 16
  const int D = in_sizes[1] / E;        // 512
  const int N = in_sizes[0] / D;        // 32768
  const int capacity = (int)((double)N * 1.2 / (double)E);  // 2457

  char* ws = (char*)d_ws;
  int* routes = (int*)ws;                       ws += (size_t)N * sizeof(int);
  float* probs = (float*)ws;                    ws += (size_t)N * sizeof(float);
  int* tok_idx = (int*)ws;                      ws += (size_t)E * capacity * sizeof(int);
  int* cnt = (int*)ws;

  router_kernel<<<N / 16, 256, 0, stream>>>(x, sw, sb, routes, probs, N);

  scan_dispatch_kernel<<<1, 1024, 0, stream>>>(routes, probs, x, tok_idx, cnt,
                                               out, N, capacity);

  const int MT = (capacity + 127) / 128;
  dim3 grid(D / 64, MT, E);
  expert_gemm_kernel<<<grid, 256, 0, stream>>>(x, ew, eb, probs, tok_idx, cnt,
                                               out, capacity);
}